// TandemIF_86096914416163
// MI455X (gfx1250) — compile-verified
//
#include <hip/hip_runtime.h>

// Integrate-and-fire scan:  for t in 0..63:  v += x[t]; s = (v >= 1); v -= s
// x_seq: (64, 128, 4096) f32 -> s_seq: (64, 128, 4096) f32.
// Memory-bound: 268 MB streamed once, ~11.5 us floor @ 23.3 TB/s.
// Strategy: register double-buffered software pipeline so each wave keeps
// G b128 NT loads in flight while computing the previous group (split
// LOADcnt/STOREcnt counters on CDNA5 let stores overlap loads for free).

typedef float vf4 __attribute__((ext_vector_type(4)));

#define IF_T 64          // timesteps (scan axis)
#define IF_G 4           // timesteps per pipeline group (loads in flight)
#define V_TH 1.0f

__global__ __launch_bounds__(256) void TandemIF_if_scan_kernel(
    const vf4* __restrict__ x,   // [T][strideV4] of float4
    vf4* __restrict__ out,       // [T][strideV4] of float4
    int strideV4)                // float4 elements per timestep (HW/4)
{
    const size_t i = (size_t)blockIdx.x * blockDim.x + threadIdx.x;
    const size_t stride = (size_t)strideV4;
    const vf4* __restrict__ xp = x + i;
    vf4* __restrict__ op = out + i;

    vf4 v = (vf4){0.0f, 0.0f, 0.0f, 0.0f};

    // Double-buffered x tiles, fully unrolled -> pure VGPRs (2*4*4 = 32 regs).
    vf4 buf[2][IF_G];

    // Prologue: issue loads for group 0.
#pragma unroll
    for (int k = 0; k < IF_G; ++k)
        buf[0][k] = __builtin_nontemporal_load(xp + (size_t)k * stride);

    constexpr int NGRP = IF_T / IF_G;   // 16 groups

#pragma unroll
    for (int g = 0; g < NGRP; ++g) {
        const int cur = g & 1;
        const int nxt = cur ^ 1;

        // Issue next group's loads before touching this group's data:
        // keeps IF_G b128 loads outstanding -> wait becomes loadcnt<=IF_G.
        if (g + 1 < NGRP) {
#pragma unroll
            for (int k = 0; k < IF_G; ++k)
                buf[nxt][k] = __builtin_nontemporal_load(
                    xp + (size_t)((g + 1) * IF_G + k) * stride);
        }

        // Deep prefetch two groups ahead (global_prefetch_b8, speculative).
        if (g + 2 < NGRP)
            __builtin_prefetch(
                (const void*)(xp + (size_t)((g + 2) * IF_G) * stride), 0, 0);

        // Consume current group: sequential IF recurrence on v.
#pragma unroll
        for (int k = 0; k < IF_G; ++k) {
            const int t = g * IF_G + k;
            v += buf[cur][k];

            vf4 spike;
            spike.x = (v.x >= V_TH) ? 1.0f : 0.0f;
            spike.y = (v.y >= V_TH) ? 1.0f : 0.0f;
            spike.z = (v.z >= V_TH) ? 1.0f : 0.0f;
            spike.w = (v.w >= V_TH) ? 1.0f : 0.0f;

            v -= spike;  // spike * V_TH with V_TH == 1.0f

            __builtin_nontemporal_store(spike, op + (size_t)t * stride);
        }
    }
}

extern "C" void kernel_launch(void* const* d_in, const int* in_sizes, int n_in,
                              void* d_out, int out_size, void* d_ws, size_t ws_size,
                              hipStream_t stream) {
    (void)n_in; (void)d_ws; (void)ws_size; (void)out_size;

    const float* x = (const float*)d_in[0];
    float* out = (float*)d_out;

    const int total  = in_sizes[0];        // 64 * 128 * 4096 = 33,554,432
    const int per_t  = total / IF_T;       // 524,288 neurons
    const int per_t4 = per_t / 4;          // 131,072 float4 lanes

    const int block = 256;                 // 8 wave32s per workgroup
    const int grid  = (per_t4 + block - 1) / block;   // 512 blocks

    TandemIF_if_scan_kernel<<<grid, block, 0, stream>>>(
        (const vf4*)x, (vf4*)out, per_t4);
}